// DeepQI_24257975288282
// MI455X (gfx1250) — compile-verified
//
#include <hip/hip_runtime.h>

#define FF   32
#define VV   10000
#define DD   496
#define HH   1024
#define BB   16384

typedef __attribute__((ext_vector_type(2))) float v2f;
typedef __attribute__((ext_vector_type(8))) float v8f;

// ---------------------------------------------------------------------------
// Kernel A: hpart[b] = sum_n relu((xv @ W1)[b,n] + b1[n]) * W2[D + n]
// One block = 4 waves, each wave owns a 16-wide slice of N per iteration.
// 16x16x4 f32 WMMA, K=32 via 8 chained accumulating WMMAs (exact f32 math).
// ---------------------------------------------------------------------------
__global__ __launch_bounds__(128) void deepqi_mlp_wmma(
    const float* __restrict__ xv, const float* __restrict__ W1,
    const float* __restrict__ b1, const float* __restrict__ W2,
    float* __restrict__ hpart)
{
    const int b0     = blockIdx.x * 16;
    const int tid    = threadIdx.x;
    const int wave   = tid >> 5;
    const int lane   = tid & 31;
    const int lane15 = lane & 15;
    const bool hi    = (lane >= 16);

    // A fragments: xv tile rows b0..b0+15, K=32.
    // Lane layout (16x4 f32 A): lanes 0-15 -> K=4kc+0,1 ; lanes 16-31 -> K=4kc+2,3
    float a0[8], a1[8];
    {
        const float* xr = xv + (b0 + lane15) * FF;
#pragma unroll
        for (int kc = 0; kc < 8; ++kc) {
            const int kb = kc * 4 + (hi ? 2 : 0);
            a0[kc] = xr[kb];
            a1[kc] = xr[kb + 1];
        }
    }

    float acc[8];
#pragma unroll
    for (int r = 0; r < 8; ++r) acc[r] = 0.0f;

    for (int iter = 0; iter < HH / 64; ++iter) {
        const int nb = iter * 64 + wave * 16;
        const int n  = nb + lane15;

        v8f c = {};
#pragma unroll
        for (int kc = 0; kc < 8; ++kc) {
            const int kb = kc * 4 + (hi ? 2 : 0);
            v2f a;  a.x  = a0[kc];            a.y  = a1[kc];
            v2f bf; bf.x = W1[kb * HH + n];   bf.y = W1[(kb + 1) * HH + n];
            c = __builtin_amdgcn_wmma_f32_16x16x4_f32(
                    false, a, false, bf, (short)0, c, false, false);
        }

        const float bias = b1[n];
        const float w2v  = W2[DD + n];
#pragma unroll
        for (int r = 0; r < 8; ++r) {
            float h = c[r] + bias;
            h = h > 0.0f ? h : 0.0f;
            acc[r] += h * w2v;
        }
    }

    // C/D layout: VGPR r holds row b0+r (lanes 0-15) and row b0+r+8 (lanes 16-31).
    __shared__ float hp[16];
    if (tid < 16) hp[tid] = 0.0f;
    __syncthreads();
    const int rofs = hi ? 8 : 0;
#pragma unroll
    for (int r = 0; r < 8; ++r) atomicAdd(&hp[r + rofs], acc[r]);
    __syncthreads();
    if (tid < 16) hpart[b0 + tid] = hp[tid];
}

// ---------------------------------------------------------------------------
// Kernel B: FM interaction (bandwidth-bound gather) + final head.
// One block per batch row; lane t owns float4 chunk t of the 496-wide row.
// out[b] = 0.5*(s^2 - ss) . W2[:D]  +  hpart[b]  +  b2
// ---------------------------------------------------------------------------
__global__ __launch_bounds__(128) void deepqi_fm(
    const float* __restrict__ xv, const int* __restrict__ xi,
    const float* __restrict__ emb, const float* __restrict__ W2,
    const float* __restrict__ b2, const float* __restrict__ hpart,
    float* __restrict__ out)
{
    const int b = blockIdx.x;
    const int t = threadIdx.x;

    __shared__ float sxv[FF];
    __shared__ int   sidx[FF];
    if (t < FF) {
        sxv[t]  = xv[b * FF + t];
        sidx[t] = xi[b * FF + t];
    }
    __syncthreads();

    float4 s  = {0.f, 0.f, 0.f, 0.f};
    float4 ss = {0.f, 0.f, 0.f, 0.f};
    if (t < DD / 4) {
#pragma unroll 4
        for (int f = 0; f < FF; ++f) {
            const float v = sxv[f];
            const float* rowp = emb + ((size_t)f * VV + (size_t)sidx[f]) * DD;
            const float4 g = *(const float4*)(rowp + t * 4);  // 16B aligned: 1984 % 16 == 0
            const float ex = g.x * v, ey = g.y * v, ez = g.z * v, ew = g.w * v;
            s.x += ex;      s.y += ey;      s.z += ez;      s.w += ew;
            ss.x += ex*ex;  ss.y += ey*ey;  ss.z += ez*ez;  ss.w += ew*ew;
        }
    }

    float partial = 0.0f;
    if (t < DD / 4) {
        float4 q;
        q.x = 0.5f * (s.x * s.x - ss.x);
        q.y = 0.5f * (s.y * s.y - ss.y);
        q.z = 0.5f * (s.z * s.z - ss.z);
        q.w = 0.5f * (s.w * s.w - ss.w);
        const float4 w = *(const float4*)(W2 + t * 4);
        partial = q.x * w.x + q.y * w.y + q.z * w.z + q.w * w.w;
    }

    // wave32 reduction, then 4-wave LDS combine
#pragma unroll
    for (int o = 16; o > 0; o >>= 1)
        partial += __shfl_down(partial, o, 32);
    __shared__ float wsum[4];
    if ((t & 31) == 0) wsum[t >> 5] = partial;
    __syncthreads();
    if (t == 0)
        out[b] = wsum[0] + wsum[1] + wsum[2] + wsum[3] + b2[0] + hpart[b];
}

// ---------------------------------------------------------------------------
extern "C" void kernel_launch(void* const* d_in, const int* in_sizes, int n_in,
                              void* d_out, int out_size, void* d_ws, size_t ws_size,
                              hipStream_t stream) {
    const float* xv  = (const float*)d_in[0];
    const int*   xi  = (const int*)  d_in[1];
    const float* emb = (const float*)d_in[2];
    const float* W1  = (const float*)d_in[3];
    const float* b1  = (const float*)d_in[4];
    const float* W2  = (const float*)d_in[5];
    const float* b2  = (const float*)d_in[6];
    float* out   = (float*)d_out;
    float* hpart = (float*)d_ws;   // B floats = 64 KB

    deepqi_mlp_wmma<<<BB / 16, 128, 0, stream>>>(xv, W1, b1, W2, hpart);
    deepqi_fm<<<BB, 128, 0, stream>>>(xv, xi, emb, W2, b2, hpart, out);
}